// JAGNNLayer_84988812853630
// MI455X (gfx1250) — compile-verified
//
#include <hip/hip_runtime.h>

typedef float v2f __attribute__((ext_vector_type(2)));
typedef float v8f __attribute__((ext_vector_type(8)));

#define OUTF 256
#define HEADS 4
#define DH 64
#define KNBR 16

// ---------------------------------------------------------------------------
// init: gat = 0, m = -inf, denom = 0
// ---------------------------------------------------------------------------
__global__ void __launch_bounds__(256) init_kernel(float* __restrict__ gat,
                                                   float* __restrict__ m,
                                                   float* __restrict__ denom,
                                                   int N) {
    int idx = blockIdx.x * blockDim.x + threadIdx.x;
    if (idx < N * OUTF) gat[idx] = 0.0f;
    if (idx < N * HEADS) {
        m[idx] = -__builtin_inff();
        denom[idx] = 0.0f;
    }
}

// ---------------------------------------------------------------------------
// C[M x 256] (+coloff, ldc) = A[M x 256] @ B[256 x 256]   (fp32 WMMA)
// One wave computes a 16x64 strip: 4 accumulators, A fragment reused 4x.
// K loop: 64 x v_wmma_f32_16x16x4_f32 per strip.
// ---------------------------------------------------------------------------
__global__ void __launch_bounds__(256)
gemm_wmma_f32(const float* __restrict__ A, const float* __restrict__ B,
              float* __restrict__ C, int ldc, int coloff, int Mtiles) {
    const int wave = threadIdx.x >> 5;
    const int lane = threadIdx.x & 31;
    const int tile = blockIdx.x * 8 + wave;
    if (tile >= Mtiles * 4) return;          // wave-uniform; full waves keep EXEC=all-1s
    const int mt  = tile >> 2;               // 16-row tile
    const int ntg = tile & 3;                // 64-col group

    const int half = lane >> 4;              // 0: lanes 0-15, 1: lanes 16-31
    const int l16  = lane & 15;
    const int colbase = ntg * 64;

    const float* Arow = A + (size_t)(mt * 16 + l16) * OUTF;

    v8f acc0 = {0,0,0,0,0,0,0,0};
    v8f acc1 = {0,0,0,0,0,0,0,0};
    v8f acc2 = {0,0,0,0,0,0,0,0};
    v8f acc3 = {0,0,0,0,0,0,0,0};

    #pragma unroll 4
    for (int k = 0; k < 256; k += 4) {
        const int ka = k + 2 * half;         // lanes 0-15: K=k,k+1 ; lanes 16-31: K=k+2,k+3
        v2f a = *(const v2f*)(Arow + ka);

        const float* Brow0 = B + (size_t)ka * OUTF + colbase + l16;
        const float* Brow1 = Brow0 + OUTF;
        v2f b0; b0.x = Brow0[0];  b0.y = Brow1[0];
        v2f b1; b1.x = Brow0[16]; b1.y = Brow1[16];
        v2f b2; b2.x = Brow0[32]; b2.y = Brow1[32];
        v2f b3; b3.x = Brow0[48]; b3.y = Brow1[48];

        acc0 = __builtin_amdgcn_wmma_f32_16x16x4_f32(false, a, false, b0, (short)0, acc0, false, false);
        acc1 = __builtin_amdgcn_wmma_f32_16x16x4_f32(false, a, false, b1, (short)0, acc1, false, false);
        acc2 = __builtin_amdgcn_wmma_f32_16x16x4_f32(false, a, false, b2, (short)0, acc2, false, false);
        acc3 = __builtin_amdgcn_wmma_f32_16x16x4_f32(false, a, false, b3, (short)0, acc3, false, false);
    }

    // C/D layout: VGPR v -> row v (lanes 0-15) / row v+8 (lanes 16-31), N = lane%16
    const int row0 = mt * 16 + half * 8;
    #pragma unroll
    for (int v = 0; v < 8; ++v) {
        float* Crow = C + (size_t)(row0 + v) * ldc + coloff + colbase + l16;
        Crow[0]  = acc0[v];
        Crow[16] = acc1[v];
        Crow[32] = acc2[v];
        Crow[48] = acc3[v];
    }
}

// ---------------------------------------------------------------------------
// el[n,h] = sum_d feat[n,h,d]*al[h,d] ; er likewise
// ---------------------------------------------------------------------------
__global__ void __launch_bounds__(256)
el_er_kernel(const float* __restrict__ feat, const float* __restrict__ al,
             const float* __restrict__ ar, float* __restrict__ el,
             float* __restrict__ er, int N) {
    int idx = blockIdx.x * blockDim.x + threadIdx.x;   // n*H + h
    if (idx >= N * HEADS) return;
    int n = idx >> 2, h = idx & 3;
    const float* f   = feat + (size_t)n * OUTF + h * DH;
    const float* alv = al + h * DH;
    const float* arv = ar + h * DH;
    float sl = 0.f, sr = 0.f;
    #pragma unroll 8
    for (int d = 0; d < DH; ++d) { float fv = f[d]; sl += fv * alv[d]; sr += fv * arv[d]; }
    el[idx] = sl; er[idx] = sr;
}

__device__ inline void atomicMaxF(float* addr, float val) {
    unsigned* u = (unsigned*)addr;
    unsigned cur = *u;
    while (val > __uint_as_float(cur)) {
        unsigned old = atomicCAS(u, cur, __float_as_uint(val));
        if (old == cur) break;
        cur = old;
    }
}

// ---------------------------------------------------------------------------
// pass1: e = leaky_relu(el[src]+er[dst]); store; segment max via atomicMax
// ---------------------------------------------------------------------------
__global__ void __launch_bounds__(256)
edge_pass1(const int* __restrict__ src, const int* __restrict__ dst,
           const float* __restrict__ el, const float* __restrict__ er,
           float* __restrict__ ebuf, float* __restrict__ m, int E) {
    int idx = blockIdx.x * blockDim.x + threadIdx.x;   // e*H + h
    if (idx >= E * HEADS) return;
    int e = idx >> 2, h = idx & 3;
    int d = dst[e];
    float v = el[src[e] * HEADS + h] + er[d * HEADS + h];
    v = v > 0.f ? v : 0.2f * v;
    ebuf[idx] = v;
    atomicMaxF(&m[d * HEADS + h], v);
}

// ---------------------------------------------------------------------------
// pass2: ex = exp(e - m[dst]); store; segment sum via atomicAdd
// ---------------------------------------------------------------------------
__global__ void __launch_bounds__(256)
edge_pass2(const int* __restrict__ dst, float* __restrict__ ebuf,
           const float* __restrict__ m, float* __restrict__ denom, int E) {
    int idx = blockIdx.x * blockDim.x + threadIdx.x;
    if (idx >= E * HEADS) return;
    int e = idx >> 2, h = idx & 3;
    float ex = expf(ebuf[idx] - m[dst[e] * HEADS + h]);
    ebuf[idx] = ex;
    atomicAdd(&denom[dst[e] * HEADS + h], ex);
}

// ---------------------------------------------------------------------------
// message pass: one wave per edge; gat[dst] += a_h * feat[src]
// ---------------------------------------------------------------------------
__global__ void __launch_bounds__(256)
edge_msg(const int* __restrict__ src, const int* __restrict__ dst,
         const float* __restrict__ ew, const float* __restrict__ ebuf,
         const float* __restrict__ denom, const float* __restrict__ feat,
         float* __restrict__ gat, int E) {
    const int wave = threadIdx.x >> 5;
    const int lane = threadIdx.x & 31;
    const int e = blockIdx.x * 8 + wave;
    if (e >= E) return;
    const int s = src[e], d = dst[e];
    const float w = ew[e];
    const float a0 = ebuf[e*HEADS+0] / denom[d*HEADS+0] * w;
    const float a1 = ebuf[e*HEADS+1] / denom[d*HEADS+1] * w;
    const float a2 = ebuf[e*HEADS+2] / denom[d*HEADS+2] * w;
    const float a3 = ebuf[e*HEADS+3] / denom[d*HEADS+3] * w;
    const float* fs = feat + (size_t)s * OUTF;
    float* gd = gat + (size_t)d * OUTF;
    #pragma unroll
    for (int i = 0; i < 8; ++i) {
        int c = lane + 32 * i;               // c>>6 == i>>1 : head constant per iter
        float ah = (i < 2) ? a0 : (i < 4) ? a1 : (i < 6) ? a2 : a3;
        atomicAdd(&gd[c], ah * fs[c]);
    }
}

// ---------------------------------------------------------------------------
// gat += bias b (broadcast over rows)
// ---------------------------------------------------------------------------
__global__ void __launch_bounds__(256)
bias_kernel(float* __restrict__ gat, const float* __restrict__ b, int N) {
    int idx = blockIdx.x * blockDim.x + threadIdx.x;
    if (idx >= N * OUTF) return;
    gat[idx] += b[idx & (OUTF - 1)];
}

// ---------------------------------------------------------------------------
// enhance select: one wave per node. Top-5 / bottom-2 by nbr_w (ties -> lowest
// index, matching jax.lax.top_k), mean-pool gathered gat rows, small 256->8
// GEMVs with cross-lane reduction. Writes t = top5@Wt+bt, bo = bot2@Wb+bb.
// ---------------------------------------------------------------------------
__global__ void __launch_bounds__(256)
enhance_select(const int* __restrict__ nbr_idx, const float* __restrict__ nbr_w,
               const float* __restrict__ gat,
               const float* __restrict__ Wt, const float* __restrict__ bt,
               const float* __restrict__ Wb, const float* __restrict__ bb,
               float* __restrict__ tout, float* __restrict__ boout, int N) {
    const int wave = threadIdx.x >> 5;
    const int lane = threadIdx.x & 31;
    const int n = blockIdx.x * 8 + wave;
    if (n >= N) return;

    float w[KNBR];
    const float* wp = nbr_w + (size_t)n * KNBR;
    #pragma unroll
    for (int i = 0; i < KNBR; ++i) w[i] = wp[i];

    int ti[5];
    unsigned mask = 0;
    #pragma unroll
    for (int j = 0; j < 5; ++j) {
        int best = 0; float bv = -__builtin_inff();
        #pragma unroll
        for (int i = 0; i < KNBR; ++i)
            if (!((mask >> i) & 1u) && w[i] > bv) { bv = w[i]; best = i; }
        ti[j] = best; mask |= 1u << best;
    }
    int bi[2];
    unsigned mask2 = 0;
    #pragma unroll
    for (int j = 0; j < 2; ++j) {
        int best = 0; float bv = __builtin_inff();
        #pragma unroll
        for (int i = 0; i < KNBR; ++i)
            if (!((mask2 >> i) & 1u) && w[i] < bv) { bv = w[i]; best = i; }
        bi[j] = best; mask2 |= 1u << best;
    }

    const int* ni = nbr_idx + (size_t)n * KNBR;
    int rows[7];
    #pragma unroll
    for (int j = 0; j < 5; ++j) rows[j] = ni[ti[j]];
    rows[5] = ni[bi[0]];
    rows[6] = ni[bi[1]];

    float tacc[8] = {0,0,0,0,0,0,0,0};
    float bacc[8] = {0,0,0,0,0,0,0,0};
    #pragma unroll
    for (int i = 0; i < 8; ++i) {
        int c = lane + 32 * i;
        float tm = (gat[(size_t)rows[0]*OUTF + c] + gat[(size_t)rows[1]*OUTF + c] +
                    gat[(size_t)rows[2]*OUTF + c] + gat[(size_t)rows[3]*OUTF + c] +
                    gat[(size_t)rows[4]*OUTF + c]) * 0.2f;
        float bm = (gat[(size_t)rows[5]*OUTF + c] + gat[(size_t)rows[6]*OUTF + c]) * 0.5f;
        #pragma unroll
        for (int j = 0; j < 8; ++j) {
            tacc[j] += tm * Wt[c * 8 + j];
            bacc[j] += bm * Wb[c * 8 + j];
        }
    }
    #pragma unroll
    for (int j = 0; j < 8; ++j) {
        float t = tacc[j], bo = bacc[j];
        #pragma unroll
        for (int off = 16; off; off >>= 1) {
            t  += __shfl_xor(t, off, 32);
            bo += __shfl_xor(bo, off, 32);
        }
        if (lane == 0) {
            tout[(size_t)n * 8 + j]  = t  + bt[j];
            boout[(size_t)n * 8 + j] = bo + bb[j];
        }
    }
}

// ---------------------------------------------------------------------------
// epilogue: out[n, off+c] += t@Wp[256:264] + bo@Wp[264:272] + bp
// ---------------------------------------------------------------------------
__global__ void __launch_bounds__(256)
proj_epilogue(const float* __restrict__ t, const float* __restrict__ bo,
              const float* __restrict__ Wp, const float* __restrict__ bp,
              float* __restrict__ out, int N, int coloff) {
    int idx = blockIdx.x * blockDim.x + threadIdx.x;
    if (idx >= N * OUTF) return;
    int n = idx >> 8, c = idx & (OUTF - 1);
    float s = bp[c];
    const float* tp  = t  + (size_t)n * 8;
    const float* bop = bo + (size_t)n * 8;
    #pragma unroll
    for (int j = 0; j < 8; ++j) s += tp[j]  * Wp[(size_t)(256 + j) * OUTF + c];
    #pragma unroll
    for (int j = 0; j < 8; ++j) s += bop[j] * Wp[(size_t)(264 + j) * OUTF + c];
    out[(size_t)n * (3 * OUTF) + coloff + c] += s;
}

// ---------------------------------------------------------------------------
extern "C" void kernel_launch(void* const* d_in, const int* in_sizes, int n_in,
                              void* d_out, int out_size, void* d_ws, size_t ws_size,
                              hipStream_t stream) {
    (void)n_in; (void)out_size; (void)ws_size;
    const int N = in_sizes[0] / OUTF;     // 30000
    const int E = in_sizes[1];            // 480000

    const float* h  = (const float*)d_in[0];
    const float* Wt = (const float*)d_in[28];
    const float* bt = (const float*)d_in[29];
    const float* Wb = (const float*)d_in[30];
    const float* bb = (const float*)d_in[31];
    const float* Wp = (const float*)d_in[32];
    const float* bp = (const float*)d_in[33];
    float* out = (float*)d_out;

    // workspace carve (all fp32, ~73 MB total)
    float* ws    = (float*)d_ws;
    float* feat  = ws;  ws += (size_t)N * OUTF;
    float* gat   = ws;  ws += (size_t)N * OUTF;
    float* el    = ws;  ws += (size_t)N * HEADS;
    float* er    = ws;  ws += (size_t)N * HEADS;
    float* mbuf  = ws;  ws += (size_t)N * HEADS;
    float* denom = ws;  ws += (size_t)N * HEADS;
    float* ebuf  = ws;  ws += (size_t)E * HEADS;
    float* tbuf  = ws;  ws += (size_t)N * 8;
    float* bobuf = ws;  ws += (size_t)N * 8;

    const int Mtiles     = N / 16;                 // 1875 (exact)
    const int gemmBlocks = (Mtiles * 4 + 7) / 8;   // 938
    const int nOutBlocks = (N * OUTF + 255) / 256;
    const int nhBlocks   = (N * HEADS + 255) / 256;
    const int ehBlocks   = (E * HEADS + 255) / 256;
    const int edgeBlocks = (E + 7) / 8;
    const int nodeBlocks = (N + 7) / 8;

    for (int et = 0; et < 3; ++et) {
        const int base = 1 + et * 9;
        const int*   src  = (const int*)d_in[base + 0];
        const int*   dst  = (const int*)d_in[base + 1];
        const float* ew   = (const float*)d_in[base + 2];
        const int*   nidx = (const int*)d_in[base + 3];
        const float* nw   = (const float*)d_in[base + 4];
        const float* W    = (const float*)d_in[base + 5];
        const float* al   = (const float*)d_in[base + 6];
        const float* ar   = (const float*)d_in[base + 7];
        const float* b    = (const float*)d_in[base + 8];

        init_kernel<<<nOutBlocks, 256, 0, stream>>>(gat, mbuf, denom, N);
        gemm_wmma_f32<<<gemmBlocks, 256, 0, stream>>>(h, W, feat, OUTF, 0, Mtiles);
        el_er_kernel<<<nhBlocks, 256, 0, stream>>>(feat, al, ar, el, er, N);
        edge_pass1<<<ehBlocks, 256, 0, stream>>>(src, dst, el, er, ebuf, mbuf, E);
        edge_pass2<<<ehBlocks, 256, 0, stream>>>(dst, ebuf, mbuf, denom, E);
        edge_msg<<<edgeBlocks, 256, 0, stream>>>(src, dst, ew, ebuf, denom, feat, gat, E);
        bias_kernel<<<nOutBlocks, 256, 0, stream>>>(gat, b, N);
        enhance_select<<<nodeBlocks, 256, 0, stream>>>(nidx, nw, gat, Wt, bt, Wb, bb,
                                                       tbuf, bobuf, N);
        gemm_wmma_f32<<<gemmBlocks, 256, 0, stream>>>(gat, Wp, out, 3 * OUTF,
                                                      et * OUTF, Mtiles);
        proj_epilogue<<<nOutBlocks, 256, 0, stream>>>(tbuf, bobuf, Wp, bp, out, N,
                                                      et * OUTF);
    }
}